// ConvDecoder_52802327937319
// MI455X (gfx1250) — compile-verified
//
#include <hip/hip_runtime.h>
#include <hip/hip_bf16.h>

#define BT    128
#define NPIX  16384   // 128*128

typedef __attribute__((ext_vector_type(2))) float v2f;
typedef __attribute__((ext_vector_type(8))) float v8f;

// Low 32 bits of a flat pointer into LDS == LDS byte address (aperture is in the
// high 32 bits on CDNA5 per ISA flat-addressing rules).
__device__ __forceinline__ unsigned lds_off32(const float* p) {
    return (unsigned)(unsigned long long)p;
}

// Exact fp32 wave(32)-wide sum, broadcast to all lanes, using two chained
// V_WMMA_F32_16X16X4_F32 ops. Relies only on the documented A(16x4) layout
// (VGPR0: K=0 lanes 0-15 / K=2 lanes 16-31; VGPR1: K=1 / K=3, M=lane%16) and
// the documented 16x16 f32 C/D layout (VGPR r: M=r lanes 0-15, M=r+8 lanes 16-31).
// Stage 1: D[m,*] = 2*(x_m + x_{m+16}).  Row-sum of each lane's 8 D VGPRs gives
// y = 2*S_half (half depends on lane<16).  Stage 2: D2[m,*] = 2*(y_m + y_{m+16})
// = 4 * total.  Requires EXEC all ones (call only from uniform control flow).
__device__ __forceinline__ float wave_sum_bcast(float x) {
    v2f a;    a[0] = x;    a[1] = x;
    v2f ones; ones[0] = 1.0f; ones[1] = 1.0f;
    v8f z = {};
    v8f d1 = __builtin_amdgcn_wmma_f32_16x16x4_f32(false, a, false, ones,
                                                   (short)0, z, false, false);
    float y = ((d1[0] + d1[1]) + (d1[2] + d1[3])) +
              ((d1[4] + d1[5]) + (d1[6] + d1[7]));
    v2f a2; a2[0] = y; a2[1] = y;
    v8f d2 = __builtin_amdgcn_wmma_f32_16x16x4_f32(false, a2, false, ones,
                                                   (short)0, z, false, false);
    return d2[0] * 0.25f;
}

__device__ __forceinline__ int clamp64(int v) {
    return v < 0 ? 0 : (v > 63 ? 63 : v);
}

// Bilinear sample (zero padding OOB) of 4 channels of one object from LDS.
__device__ __forceinline__ void sample4(const float* __restrict__ T, int obj,
                                        float xs, float ys, float out[4]) {
    float xf = floorf(xs), yf = floorf(ys);
    int   ix = (int)xf,    iy = (int)yf;
    float fx = xs - xf,    fy = ys - yf;
    float gx0 = 1.0f - fx, gy0 = 1.0f - fy;
    float w00 = gx0 * gy0, w10 = fx * gy0, w01 = gx0 * fy, w11 = fx * fy;
    bool vx0 = (unsigned)ix       < 64u;
    bool vx1 = (unsigned)(ix + 1) < 64u;
    bool vy0 = (unsigned)iy       < 64u;
    bool vy1 = (unsigned)(iy + 1) < 64u;
    w00 = (vx0 && vy0) ? w00 : 0.0f;
    w10 = (vx1 && vy0) ? w10 : 0.0f;
    w01 = (vx0 && vy1) ? w01 : 0.0f;
    w11 = (vx1 && vy1) ? w11 : 0.0f;
    int cx0 = clamp64(ix), cx1 = clamp64(ix + 1);
    int cy0 = clamp64(iy), cy1 = clamp64(iy + 1);
    int o00 = cy0 * 64 + cx0, o10 = cy0 * 64 + cx1;
    int o01 = cy1 * 64 + cx0, o11 = cy1 * 64 + cx1;
    const float* P = T + obj * 4 * 4096;
#pragma unroll
    for (int c = 0; c < 4; ++c) {
        out[c] = w00 * P[o00] + w10 * P[o10] + w01 * P[o01] + w11 * P[o11];
        P += 4096;
    }
}

// smem layout (floats): [0,32768)   : 8 planes (obj0:{t0,t1,t2,cont}, obj1:{...}), 64x64 each
//                       [32768,33024): BX[2][128]   [33024,33280): BY[2][128]
__global__ __launch_bounds__(256)
void convdec_kernel(const float* __restrict__ x,
                    const float* __restrict__ logsigma,
                    const float* __restrict__ templ,
                    const float* __restrict__ cont,
                    float* __restrict__ out) {
    __shared__ float smem[33280];
    const int tid = threadIdx.x;
    const float sigma = expf(logsigma[0]);

    // ---- Stage templates into LDS with CDNA5 async copies (128 KB) ----
    // Plane p = o*4+c ; c<3 -> templ[o][c], c==3 -> cont[o].
#pragma unroll 1
    for (int k = 0; k < 32; ++k) {
        int plane  = k >> 2;          // 0..7
        int o      = plane >> 2;      // 0..1
        int c      = plane & 3;       // 0..3
        int within = ((k & 3) << 10) | (tid << 2);   // 0..4092, 4 floats per lane
        const float* src = (c < 3) ? (templ + (((o * 3 + c) << 12) + within))
                                   : (cont  + ((o << 12) + within));
        unsigned dst = lds_off32(&smem[(plane << 12) + within]);
        asm volatile("global_load_async_to_lds_b128 %0, %1, off"
                     :: "v"(dst), "v"(src) : "memory");
    }
    asm volatile("s_wait_asynccnt 0" ::: "memory");
    __syncthreads();

    // ---- In-place transform: templ += 5, cont -> sigmoid ----
#pragma unroll 1
    for (int k = 0; k < 32; ++k) {
        int plane = k >> 2;
        int c     = plane & 3;
        int idx   = (plane << 12) | ((k & 3) << 10) | (tid << 2);
        if (c < 3) {
#pragma unroll
            for (int j = 0; j < 4; ++j) smem[idx + j] += 5.0f;
        } else {
#pragma unroll
            for (int j = 0; j < 4; ++j) {
                float v = smem[idx + j];
                smem[idx + j] = 1.0f / (1.0f + expf(-v));
            }
        }
    }
    // ---- Affine offsets per (obj, bt): xs = 32*sigma*gx + BX ----
    {
        int o  = tid >> 7;            // 0..1
        int bt = tid & 127;
        const float* px = x + ((((bt << 2) | o) << 1));
        float x0 = px[0], x1 = px[1];
        smem[32768 + (o << 7) + bt] = fmaf(0.5f * sigma, 1.5f - x0, 31.5f);
        smem[33024 + (o << 7) + bt] = fmaf(0.5f * sigma, 1.5f - x1, 31.5f);
    }
    __syncthreads();

    const float a   = 32.0f * sigma;
    const int wv    = tid >> 5;
    const int lane  = tid & 31;
    const float* BX0 = smem + 32768;
    const float* BX1 = smem + 32768 + 128;
    const float* BY0 = smem + 33024;
    const float* BY1 = smem + 33024 + 128;

#pragma unroll 1
    for (int j = 0; j < 8; ++j) {
        int p    = (blockIdx.x << 6) + (wv << 3) + j;   // pixel index, 256 blocks * 64
        int hrow = p >> 7, wcol = p & 127;
        float gx = (wcol + 0.5f) * (1.0f / 64.0f) - 1.0f;
        float gy = (hrow + 0.5f) * (1.0f / 64.0f) - 1.0f;

        // Pass 1: object 0 samples, exp, per-lane partial denominators (4 bt/lane)
        float part0 = 0.f, part1 = 0.f, part2 = 0.f, part3 = 0.f;
        float eA[4][4];
#pragma unroll
        for (int i = 0; i < 4; ++i) {
            int bt = (lane << 2) | i;
            float xs = fmaf(a, gx, BX0[bt]);
            float ys = fmaf(a, gy, BY0[bt]);
            float vA[4];
            sample4(smem, 0, xs, ys, vA);
            float e0 = expf(vA[0] - 5.0f); eA[i][0] = e0; part0 += e0;
            float e1 = expf(vA[1] - 5.0f); eA[i][1] = e1; part1 += e1;
            float e2 = expf(vA[2] - 5.0f); eA[i][2] = e2; part2 += e2;
            float e3 = expf(vA[3] - 5.0f); eA[i][3] = e3; part3 += e3;
        }
        // Wave-wide softmax denominators via WMMA reduction (uniform control flow)
        float rd0 = 1.0f / wave_sum_bcast(part0);
        float rd1 = 1.0f / wave_sum_bcast(part1);
        float rd2 = 1.0f / wave_sum_bcast(part2);
        float rd3 = 1.0f / wave_sum_bcast(part3);

        // Pass 2: object 1 samples, combine, store
#pragma unroll
        for (int i = 0; i < 4; ++i) {
            int bt = (lane << 2) | i;
            float xs = fmaf(a, gx, BX1[bt]);
            float ys = fmaf(a, gy, BY1[bt]);
            float vB[4];
            sample4(smem, 1, xs, ys, vB);
            float val = 3.0f * (eA[i][0] * rd0 * vB[0] +
                                eA[i][1] * rd1 * vB[1] +
                                eA[i][2] * rd2 * vB[2]) +
                        eA[i][3] * rd3 * vB[3];
            out[bt * NPIX + p] = val;
        }
    }
}

extern "C" void kernel_launch(void* const* d_in, const int* in_sizes, int n_in,
                              void* d_out, int out_size, void* d_ws, size_t ws_size,
                              hipStream_t stream) {
    const float* x        = (const float*)d_in[0];
    const float* logsigma = (const float*)d_in[1];
    const float* templ    = (const float*)d_in[2];
    const float* cont     = (const float*)d_in[3];
    // d_in[4] (background) is dead code in the reference.
    float* out = (float*)d_out;
    dim3 grid(256), block(256);
    hipLaunchKernelGGL(convdec_kernel, grid, block, 0, stream,
                       x, logsigma, templ, cont, out);
}